// GCN_76630806495639
// MI455X (gfx1250) — compile-verified
//
#include <hip/hip_runtime.h>
#include <math.h>

#define T_LEN   131072
#define NBATCH  8
#define WIN     4096
#define HALO    2048
#define TILE    2048
#define NTILES  (T_LEN / TILE)
#define NLAYERS 18
#define BLOCK   512
#define CHUNKS  16            // chunks of 16 time-cols per wave: WIN / (BLOCK/32) / 16
#define XB0     0
#define XB1     (8*WIN)
#define WOFF    (16*WIN)      // 65536 floats
#define LSTRIDE 536
#define AW_OFF  0             // 24x16 conv A, [p][m]
#define CB_OFF  384           // 16 conv bias
#define AM_OFF  400           // 8x16 mix A (rows: mw^T, mix_w slice, zeros), [p][m]
#define MB_OFF  528           // 8 mix bias
#define BLOB_FLOATS (NLAYERS*LSTRIDE + 1)     // 9649
#define LDS_FLOATS  (WOFF + BLOB_FLOATS)      // 75185
#define LDS_BYTES   (LDS_FLOATS * 4)          // 300740 B <= 320KB

typedef float v2f __attribute__((ext_vector_type(2)));
typedef float v8f __attribute__((ext_vector_type(8)));

static __device__ __forceinline__ v8f wmma4(v2f a, v2f b, v8f c) {
  // D(16x16,f32) = A(16x4,f32) * B(4x16,f32) + C
  return __builtin_amdgcn_wmma_f32_16x16x4_f32(false, a, false, b,
                                               (short)0, c, false, false);
}

// CDNA5 hardware V_TANH_F32 (single TRANS op; co-executes with WMMA).
static __device__ __forceinline__ float fast_tanh(float x) {
#if __has_builtin(__builtin_amdgcn_tanhf)
  return __builtin_amdgcn_tanhf(x);
#elif __has_builtin(__builtin_amdgcn_tanh_f32)
  return __builtin_amdgcn_tanh_f32(x);
#else
  return tanhf(x);
#endif
}

struct Ptrs { const float* p[75]; };

// Pre-transpose all layer params into a WMMA-ready blob in d_ws.
__global__ void GCN_pack_kernel(Ptrs P, float* __restrict__ blob) {
  const int tid = threadIdx.x; // 256 threads
  for (int l = 0; l < NLAYERS; ++l) {
    const float* cw   = P.p[1 + 4*l];   // (16, in_ch, 3)
    const float* cb   = P.p[2 + 4*l];   // (16,)
    const float* mw   = P.p[3 + 4*l];   // (8, 8)
    const float* mb   = P.p[4 + 4*l];   // (8,)
    const float* mixw = P.p[73];        // (1, 144)
    float* bl = blob + l*LSTRIDE;
    // Conv A: rows p = k*8 + c (k=tap, c=in_ch), cols m = out_ch. A[p][m]=cw[m][c][k]
    for (int r = tid; r < 384; r += 256) {
      int p = r >> 4, m = r & 15;
      int k = p >> 3, c = p & 7;
      float v;
      if (l == 0) v = (c == 0) ? cw[m*3 + k] : 0.f;   // in_ch=1, channels 1..7 zero
      else        v = cw[(m*8 + c)*3 + k];
      bl[AW_OFF + r] = v;
    }
    for (int r = tid; r < 16; r += 256) bl[CB_OFF + r] = cb[r];
    // Mix A (8 K-rows x 16 M): rows m<8 -> mw[m][p]; m==8 -> mix_w slice; else 0
    for (int r = tid; r < 128; r += 256) {
      int p = r >> 4, m = r & 15;
      float v = (m < 8) ? mw[m*8 + p] : ((m == 8) ? mixw[l*8 + p] : 0.f);
      bl[AM_OFF + r] = v;
    }
    for (int r = tid; r < 8; r += 256) bl[MB_OFF + r] = mb[r];
  }
  if (tid == 0) blob[NLAYERS*LSTRIDE] = P.p[74][0];  // mix_b
}

__global__ __launch_bounds__(BLOCK, 1)
void GCN_wavenet_kernel(const float* __restrict__ x,
                        const float* __restrict__ blob,
                        float* __restrict__ out) {
  extern __shared__ float lds[];
  const int tid    = threadIdx.x;
  const int batch  = blockIdx.y;
  const int tile   = blockIdx.x;
  const int gstart = tile * TILE - HALO;   // global time of window position q=0

  // Stage weights blob into LDS
  for (int i = tid; i < BLOB_FLOATS; i += BLOCK) lds[WOFF + i] = blob[i];
  // Stage input window, replicated across all 8 channels (layer-0 residual broadcast)
  for (int i = tid; i < WIN; i += BLOCK) {
    int g = gstart + i;
    float v = (g >= 0) ? x[(size_t)batch * T_LEN + g] : 0.f;
#pragma unroll
    for (int c = 0; c < 8; ++c) lds[XB0 + c*WIN + i] = v;
  }
  __syncthreads();

  const int  lane  = tid & 31;
  const int  wave  = tid >> 5;           // 0..15
  const int  m     = lane & 15;          // WMMA row/col index
  const bool hi    = lane >= 16;
  const int  qm    = wave * (CHUNKS * 16) + m;
  const float mixb = lds[WOFF + NLAYERS*LSTRIDE];

  float skipacc[CHUNKS];
#pragma unroll
  for (int ch = 0; ch < CHUNKS; ++ch) skipacc[ch] = 0.f;

  for (int l = 0; l < NLAYERS; ++l) {
    const int d    = 1 << ((l < 9) ? l : (l - 9));
    const int pad2 = 2 * d;
    const int wl   = WOFF + l * LSTRIDE;
    const int xcur = (l & 1) ? XB1 : XB0;
    const int xnxt = (l & 1) ? XB0 : XB1;

    // Conv A (16x24) into regs: lane m = M row; VGPR pair per K-slice j
    v2f A[6];
#pragma unroll
    for (int j = 0; j < 6; ++j) {
      int p0 = 4*j + (hi ? 2 : 0);
      A[j].x = lds[wl + AW_OFF + p0*16 + m];
      A[j].y = lds[wl + AW_OFF + (p0 + 1)*16 + m];
    }
    v8f cbv;
#pragma unroll
    for (int r = 0; r < 8; ++r) cbv[r] = lds[wl + CB_OFF + (hi ? r + 8 : r)];
    v2f AM[2];
#pragma unroll
    for (int j = 0; j < 2; ++j) {
      int p0 = 4*j + (hi ? 2 : 0);
      AM[j].x = lds[wl + AM_OFF + p0*16 + m];
      AM[j].y = lds[wl + AM_OFF + (p0 + 1)*16 + m];
    }
    float mbv[8];
#pragma unroll
    for (int r = 0; r < 8; ++r) mbv[r] = lds[wl + MB_OFF + r];

#pragma unroll
    for (int ch = 0; ch < CHUNKS; ++ch) {
      const int q = qm + ch * 16;        // window column of this lane
      const int g = gstart + q;          // global time

      // ---- dilated conv: D(16x16) = A(16x24) * Xpatch(24x16) + cb : 6 WMMAs
      v8f yv = cbv;
#pragma unroll
      for (int j = 0; j < 6; ++j) {
        int p0 = 4*j + (hi ? 2 : 0);
        v2f bv;
        {
          int k = p0 >> 3, c = p0 & 7;
          int qq = q - (2 - k) * d;
          int qc = qq < 0 ? 0 : qq;
          float v = lds[xcur + c*WIN + qc];
          bv.x = (qq < 0) ? 0.f : v;
        }
        {
          int p1 = p0 + 1;
          int k = p1 >> 3, c = p1 & 7;
          int qq = q - (2 - k) * d;
          int qc = qq < 0 ? 0 : qq;
          float v = lds[xcur + c*WIN + qc];
          bv.y = (qq < 0) ? 0.f : v;
        }
        yv = wmma4(A[j], bv, yv);
      }

      // ---- gate: lanes<16 hold tanh half (rows 0-7), lanes>=16 sigmoid half (rows 8-15)
      //      sigmoid(x) = 0.5*(1 + tanh(x/2)); z duplicated across lane halves via shfl
      const float zmask = (g >= pad2) ? 1.0f : 0.0f;  // reference zero-pads z, not x
      const float preml = hi ? 0.5f : 1.0f;
      float z[8];
#pragma unroll
      for (int r = 0; r < 8; ++r) {
        float t = fast_tanh(preml * yv[r]);
        float a = hi ? fmaf(0.5f, t, 0.5f) : t;
        float o = __shfl_xor(a, 16, 32);
        z[r] = (a * o) * zmask;                       // z[r][t], both halves
      }

      // ---- mix + residual + skip: D = [mw ; mix_w ; 0](16x8) * z(8x16) + C : 2 WMMAs
      //      C rows 0-7 = mb + x_old (residual); C row 8 = running skip accumulator
      v8f cm;
#pragma unroll
      for (int r = 0; r < 8; ++r) {
        float xo = lds[xcur + r*WIN + q];
        cm[r] = hi ? ((r == 0) ? skipacc[ch] : 0.f) : (mbv[r] + xo);
      }
#pragma unroll
      for (int j = 0; j < 2; ++j) {
        v2f bv;
        bv.x = hi ? z[4*j + 2] : z[4*j + 0];
        bv.y = hi ? z[4*j + 3] : z[4*j + 1];
        cm = wmma4(AM[j], bv, cm);
      }

      // lanes<16: rows 0-7 = next residual stream -> other LDS buffer
      if (!hi) {
#pragma unroll
        for (int r = 0; r < 8; ++r)
          lds[xnxt + r*WIN + q] = (g < 0) ? 0.f : cm[r];
      }
      // lanes>=16: row 8 = updated skip accumulator (junk on lo lanes, never read there)
      skipacc[ch] = cm[0];
    }
    __syncthreads();
  }

  // ---- output: skip + mix_b, only the non-halo half of the window
  if (hi) {
#pragma unroll
    for (int ch = 0; ch < CHUNKS; ++ch) {
      int q = qm + ch * 16;
      if (q >= HALO) {
        int g = gstart + q;
        out[(size_t)batch * T_LEN + g] = skipacc[ch] + mixb;
      }
    }
  }
}

extern "C" void kernel_launch(void* const* d_in, const int* in_sizes, int n_in,
                              void* d_out, int out_size, void* d_ws, size_t ws_size,
                              hipStream_t stream) {
  (void)in_sizes; (void)out_size; (void)ws_size;
  Ptrs P;
  for (int i = 0; i < 75; ++i) P.p[i] = (i < n_in) ? (const float*)d_in[i] : nullptr;
  float* blob = (float*)d_ws;

  GCN_pack_kernel<<<1, 256, 0, stream>>>(P, blob);

  const float* x = (const float*)d_in[0];
  dim3 grid(NTILES, NBATCH);
  GCN_wavenet_kernel<<<grid, BLOCK, LDS_BYTES, stream>>>(x, blob, (float*)d_out);
}